// PatchEmbed_21552145891782
// MI455X (gfx1250) — compile-verified
//
#include <hip/hip_runtime.h>
#include <hip/hip_bf16.h>
#include <stdint.h>

// Problem constants (from reference): x:(16,64,256,256) fp32, patch=2
#define N_    16
#define C_    64
#define H_    256
#define W_    256
#define HW_   (H_ * W_)                         // 65536
#define CHW_  ((size_t)C_ * HW_)                // 4,194,304 floats
#define HB_   128                               // H / p
#define WB_   128                               // W / p
#define RL_   256                               // C * p * p (output row length)
#define NP_   (HB_ * WB_)                       // 16384 patches
#define OUT_FLOATS ((size_t)N_ * NP_ * RL_)     // 67,108,864

// Native vector types (usable with __builtin_nontemporal_store, unlike the
// HIP_vector_type wrappers).
typedef float v4f __attribute__((ext_vector_type(4)));
typedef float v2f __attribute__((ext_vector_type(2)));

// LDS: 64 rows (ph*32 + jl) x 65 float2 (64 used + 1 pad) -> pitch 130 dwords,
// 130 mod 64 == 2  => conflict-free for both async b64 writes (lanes vary jl)
// and ds_load_b64 reads (lanes vary c).
#define LDS_PITCH2 65

__global__ __launch_bounds__(256) void patch_embed_p2_kernel(const float* __restrict__ x,
                                                             float* __restrict__ out) {
    __shared__ v2f lds2[64 * LDS_PITCH2];   // 33,280 bytes

    const uint32_t bid  = blockIdx.x;
    const uint32_t t    = threadIdx.x;
    const uint32_t lane = t & 31u;             // wave32
    const uint32_t wave = t >> 5;              // 0..7

    // block decode: bid = ((n * 128) + i) * 4 + jb
    const uint32_t n   = bid >> 9;             // / 512
    const uint32_t rem = bid & 511u;
    const uint32_t i   = rem >> 2;             // H-block index, 0..127
    const uint32_t jb  = rem & 3u;             // j-block (32 j's each), 0..3

    // ------------------------------------------------------------------
    // Load phase: async global -> LDS, 8-byte granules.
    // Lane -> jl (j within block): 32 lanes x 8B = 256B contiguous per op.
    // Each wave covers 16 (c,ph) rows; 16 async ops per thread.
    // ------------------------------------------------------------------
    const float* gbase = x + (size_t)n * CHW_
                           + (size_t)(2u * i) * W_
                           + (size_t)(jb * 64u)        // w start of j-block
                           + (size_t)(lane * 2u);      // w = 2*(jb*32+jl)
#pragma unroll
    for (int s = 0; s < 16; ++s) {
        const uint32_t row = wave * 16u + (uint32_t)s; // 0..127
        const uint32_t c   = row >> 1;
        const uint32_t ph  = row & 1u;
        const float* gp = gbase + (size_t)c * HW_ + (size_t)ph * W_;
        const uint32_t ldsOff =
            (uint32_t)(uintptr_t)&lds2[(ph * 32u + lane) * LDS_PITCH2 + c];
        asm volatile("global_load_async_to_lds_b64 %0, %1, off"
                     :: "v"(ldsOff), "v"(gp)
                     : "memory");
    }

    // Wait for this wave's async transfers, then sync the workgroup so every
    // thread sees the fully populated LDS tile.
    asm volatile("s_wait_asynccnt 0x0" ::: "memory");
    __syncthreads();

    // ------------------------------------------------------------------
    // Store phase: lane -> c for coalescing. Each thread assembles one
    // output v4f (c, ph0:pw0..1, ph1:pw0..1) from two conflict-free
    // ds_load_b64 reads, then nontemporal global_store_b128.
    // 32 lanes x 16B = 512B contiguous per wave-iteration.
    // ------------------------------------------------------------------
    float* obase = out + (size_t)n * NP_ * RL_;
#pragma unroll
    for (int s = 0; s < 8; ++s) {
        const uint32_t c  = ((uint32_t)(s >> 2)) * 32u + lane;   // 0..63
        const uint32_t jl = wave * 4u + (uint32_t)(s & 3);       // 0..31
        const v2f a = lds2[jl * LDS_PITCH2 + c];                 // ph = 0
        const v2f b = lds2[(32u + jl) * LDS_PITCH2 + c];         // ph = 1
        const uint32_t j = jb * 32u + jl;
        const uint32_t k = j * 128u + i;                         // k = j*hb + i
        v4f v;
        v.x = a.x; v.y = a.y; v.z = b.x; v.w = b.y;
        __builtin_nontemporal_store(
            v, (v4f*)(obase + (size_t)k * RL_ + (size_t)(c * 4u)));
    }

    // ori_shape tail: 4 int32 values appended after the float payload.
    if (bid == 0 && t < 4) {
        const int shp[4] = {N_, C_, H_, W_};
        ((int*)out)[OUT_FLOATS + t] = shp[t];
    }
}

extern "C" void kernel_launch(void* const* d_in, const int* in_sizes, int n_in,
                              void* d_out, int out_size, void* d_ws, size_t ws_size,
                              hipStream_t stream) {
    (void)in_sizes; (void)n_in; (void)out_size; (void)d_ws; (void)ws_size;
    const float* x = (const float*)d_in[0];
    float* out = (float*)d_out;
    // blocks: N(16) * hb(128) * (wb/32 = 4) = 8192, 256 threads (8 wave32)
    dim3 grid(N_ * HB_ * (WB_ / 32));
    patch_embed_p2_kernel<<<grid, 256, 0, stream>>>(x, out);
}